// SequenceSummary_15410342658382
// MI455X (gfx1250) — compile-verified
//
#include <hip/hip_runtime.h>
#include <math.h>

// ---- Problem constants (match setup_inputs in the reference) ----
constexpr int B   = 16;
constexpr int S   = 2048;
constexpr int D   = 1024;
constexpr int H   = 16;
constexpr int HD  = 64;              // head dim
constexpr int P   = 8;               // pma_count
constexpr int RC  = 64;              // recent_count
constexpr int CLS = 1;               // cls_count
constexpr int T   = CLS + P + RC;    // 73 summary tokens
constexpr int PH  = P * H;           // 128 attention query rows
constexpr float SCALE = 0.125f;      // 1/sqrt(HD)

typedef __attribute__((ext_vector_type(2))) float v2f;
typedef __attribute__((ext_vector_type(8))) float v8f;

// ---------------- block reductions (wave32) ----------------
__device__ __forceinline__ float blockReduceSum(float v) {
    __shared__ float sm[32];
    __syncthreads();
    int lane = threadIdx.x & 31, wid = threadIdx.x >> 5;
#pragma unroll
    for (int o = 16; o; o >>= 1) v += __shfl_down(v, o, 32);
    if (!lane) sm[wid] = v;
    __syncthreads();
    float r = 0.f;
    if (threadIdx.x < (blockDim.x >> 5)) r = sm[threadIdx.x];
    if (!wid) {
#pragma unroll
        for (int o = 16; o; o >>= 1) r += __shfl_down(r, o, 32);
        if (!lane) sm[0] = r;
    }
    __syncthreads();
    return sm[0];
}

__device__ __forceinline__ float blockReduceMax(float v) {
    __shared__ float sm[32];
    __syncthreads();
    int lane = threadIdx.x & 31, wid = threadIdx.x >> 5;
#pragma unroll
    for (int o = 16; o; o >>= 1) v = fmaxf(v, __shfl_down(v, o, 32));
    if (!lane) sm[wid] = v;
    __syncthreads();
    float r = -3.4e38f;
    if (threadIdx.x < (blockDim.x >> 5)) r = sm[threadIdx.x];
    if (!wid) {
#pragma unroll
        for (int o = 16; o; o >>= 1) r = fmaxf(r, __shfl_down(r, o, 32));
        if (!lane) sm[0] = r;
    }
    __syncthreads();
    return sm[0];
}

// ---------------- generic fp32 WMMA GEMM (compile-time strides) ----------------
// C[batch][m][n] = sum_k A[batch][m][k] * Bm[batch][k][n]
// One wave32 computes a 16x64 C tile: 4 accumulators along N sharing each A
// fragment (4 independent WMMA dep chains -> loads overlap WMMAs).
// K-stride==1 operands are loaded as 8-byte v2f (-> global_load_b64).
// A frag (16x4, ISA 7.12.2): lanes 0-15 carry K=0,1; lanes 16-31 carry K=2,3.
// C/D frag: vgpr r = row m0 + r + 8*(lane>=16), col n0 + (lane&15).
template <int SAM, int SAK, int SBK, int SBN, int SCM>
__global__ __launch_bounds__(128) void wmma_gemm_f32(
    const float* __restrict__ A, const float* __restrict__ Bm, float* __restrict__ C,
    int N, int K, long bsA, long bsB, long bsC)
{
    const int wave  = threadIdx.x >> 5;
    const int lane  = threadIdx.x & 31;
    const int half  = lane >> 4;
    const int l16   = lane & 15;
    const int nBase = (blockIdx.x * 4 + wave) * 64;   // 4 x 16 cols per wave
    const int m0    = blockIdx.y * 16;
    const int batch = blockIdx.z;
    if (nBase >= N) return;                            // wave-uniform guard

    const float* Ap = A  + (long)batch * bsA + (long)(m0 + l16) * SAM + (long)(2 * half) * SAK;
    const float* Bp = Bm + (long)batch * bsB + (long)(2 * half) * SBK + (long)(nBase + l16) * SBN;

    v8f acc0 = {0.f,0.f,0.f,0.f,0.f,0.f,0.f,0.f};
    v8f acc1 = acc0, acc2 = acc0, acc3 = acc0;

    for (int k = 0; k < K; k += 16) {
        // pull the streaming operand ~2 chunks ahead (global_prefetch_b8)
        __builtin_prefetch(Bp + (long)32 * SBK, 0, 1);
#pragma unroll
        for (int kk = 0; kk < 4; ++kk) {
            v2f a, b0, b1, b2, b3;
            if (SAK == 1) a = *(const v2f*)Ap;
            else { a.x = Ap[0]; a.y = Ap[SAK]; }
            if (SBK == 1) {
                b0 = *(const v2f*)(Bp);
                b1 = *(const v2f*)(Bp + (long)16 * SBN);
                b2 = *(const v2f*)(Bp + (long)32 * SBN);
                b3 = *(const v2f*)(Bp + (long)48 * SBN);
            } else {
                b0.x = Bp[0];                 b0.y = Bp[SBK];
                b1.x = Bp[(long)16 * SBN];    b1.y = Bp[(long)16 * SBN + SBK];
                b2.x = Bp[(long)32 * SBN];    b2.y = Bp[(long)32 * SBN + SBK];
                b3.x = Bp[(long)48 * SBN];    b3.y = Bp[(long)48 * SBN + SBK];
            }
            acc0 = __builtin_amdgcn_wmma_f32_16x16x4_f32(false, a, false, b0, (short)0, acc0, false, false);
            acc1 = __builtin_amdgcn_wmma_f32_16x16x4_f32(false, a, false, b1, (short)0, acc1, false, false);
            acc2 = __builtin_amdgcn_wmma_f32_16x16x4_f32(false, a, false, b2, (short)0, acc2, false, false);
            acc3 = __builtin_amdgcn_wmma_f32_16x16x4_f32(false, a, false, b3, (short)0, acc3, false, false);
            Ap += 4 * SAK; Bp += 4 * SBK;
        }
    }
    float* Cp = C + (long)batch * bsC + (long)(m0 + 8 * half) * SCM + (nBase + l16);
#pragma unroll
    for (int r = 0; r < 8; ++r) {
        Cp[(long)r * SCM]      = acc0[r];
        Cp[(long)r * SCM + 16] = acc1[r];
        Cp[(long)r * SCM + 32] = acc2[r];
        Cp[(long)r * SCM + 48] = acc3[r];
    }
}

// ---------------- small kernels ----------------
// q[p,n] = sum_k queries[p,k] * Wq[n,k] + bq[n]   (Wq = in_proj_w rows [0,D))
__global__ void qproj_kernel(const float* __restrict__ queries,
                             const float* __restrict__ w, const float* __restrict__ bias,
                             float* __restrict__ q) {
    int idx = blockIdx.x * blockDim.x + threadIdx.x;
    if (idx >= P * D) return;
    int p = idx / D, n = idx % D;
    const float* qr = queries + (long)p * D;
    const float* wr = w + (long)n * D;
    float acc = bias[n];
    for (int k = 0; k < D; ++k) acc += qr[k] * wr[k];
    q[idx] = acc;
}

// qk_eff[m= p*H+h, d] = sum_{j<HD} q[p, h*HD+j] * Wk[h*HD+j, d]
__global__ void qkeff_kernel(const float* __restrict__ q,
                             const float* __restrict__ in_proj_w,
                             float* __restrict__ qk_eff) {
    int idx = blockIdx.x * blockDim.x + threadIdx.x;
    if (idx >= PH * D) return;
    int m = idx / D, d = idx % D;
    int p = m / H, h = m % H;
    const float* qr = q + (long)p * D + h * HD;
    const float* wk = in_proj_w + ((long)D + h * HD) * D + d;   // Wk rows [D, 2D)
    float acc = 0.f;
    for (int j = 0; j < HD; ++j) acc += qr[j] * wk[(long)j * D];
    qk_eff[idx] = acc;
}

// total[b] = popcount(valid_mask[b, 1:S])
__global__ void totals_kernel(const unsigned char* __restrict__ vm, int* __restrict__ total) {
    int b = blockIdx.x;
    float s = 0.f;
    for (int i = 1 + threadIdx.x; i < S; i += blockDim.x) s += vm[(long)b * S + i] ? 1.f : 0.f;
    float t = blockReduceSum(s);
    if (threadIdx.x == 0) total[b] = (int)(t + 0.5f);
}

// in-place softmax over s of SCALE*scores with -1e9 masking
__global__ void softmax_kernel(float* __restrict__ scores, const unsigned char* __restrict__ vm) {
    int row = blockIdx.x;                 // row = b*PH + m
    int b = row / PH;
    float* base = scores + (long)row * S;
    const unsigned char* mb = vm + (long)b * S;
    float x[S / 256];
#pragma unroll
    for (int i = 0; i < S / 256; ++i) {
        int s = threadIdx.x + i * 256;
        x[i] = mb[s] ? base[s] * SCALE : -1e9f;
    }
    float mx = -3.4e38f;
#pragma unroll
    for (int i = 0; i < S / 256; ++i) mx = fmaxf(mx, x[i]);
    mx = blockReduceMax(mx);
    float se = 0.f;
#pragma unroll
    for (int i = 0; i < S / 256; ++i) { x[i] = expf(x[i] - mx); se += x[i]; }
    se = blockReduceSum(se);
    float inv = 1.f / se;
#pragma unroll
    for (int i = 0; i < S / 256; ++i) base[threadIdx.x + i * 256] = x[i] * inv;
}

// ctx[b,p, h*HD+i] = Wv[h*HD+i,:] . u[b, p*H+h, :] + bv[h*HD+i]   (Wv = in_proj_w rows [2D,3D))
__global__ void ctx_kernel(const float* __restrict__ u, const float* __restrict__ in_proj_w,
                           const float* __restrict__ in_proj_b, float* __restrict__ ctx) {
    int idx = blockIdx.x * blockDim.x + threadIdx.x;
    if (idx >= B * P * D) return;
    int b = idx / (P * D), r = idx % (P * D);
    int p = r / D, col = r % D, h = col >> 6;
    const float* wr = in_proj_w + ((long)2 * D + col) * D;
    const float* ur = u + ((long)b * PH + p * H + h) * D;
    float acc = in_proj_b[2 * D + col];
    for (int d = 0; d < D; ++d) acc += wr[d] * ur[d];
    ctx[idx] = acc;
}

// pooled[b,p,n] = out_w[n,:] . ctx[b,p,:] + out_b[n]
__global__ void pooled_kernel(const float* __restrict__ ctx, const float* __restrict__ out_w,
                              const float* __restrict__ out_b, float* __restrict__ pooled) {
    int idx = blockIdx.x * blockDim.x + threadIdx.x;
    if (idx >= B * P * D) return;
    int row = idx / D, n = idx % D;
    const float* wr = out_w + (long)n * D;
    const float* cr = ctx + (long)row * D;
    float acc = out_b[n];
    for (int d = 0; d < D; ++d) acc += wr[d] * cr[d];
    pooled[idx] = acc;
}

// pma[row,:] = LN(queries[p] + pooled[row]) * g + b    (row = b*P + p)
__global__ void pma_ln_kernel(const float* __restrict__ queries, const float* __restrict__ pooled,
                              const float* __restrict__ g, const float* __restrict__ bb,
                              float* __restrict__ pma) {
    int row = blockIdx.x, p = row % P;
    const float* qr = queries + (long)p * D;
    const float* pr = pooled + (long)row * D;
    float s = 0.f;
    for (int d = threadIdx.x; d < D; d += blockDim.x) s += qr[d] + pr[d];
    float mu = blockReduceSum(s) * (1.f / D);
    float s2 = 0.f;
    for (int d = threadIdx.x; d < D; d += blockDim.x) {
        float t = qr[d] + pr[d] - mu; s2 += t * t;
    }
    float var = blockReduceSum(s2) * (1.f / D);
    float rs = rsqrtf(var + 1e-5f);
    for (int d = threadIdx.x; d < D; d += blockDim.x)
        pma[(long)row * D + d] = (qr[d] + pr[d] - mu) * rs * g[d] + bb[d];
}

// summary assembly: cls token, pma tokens, right-aligned recent-64 gather
__global__ void summary_kernel(const float* __restrict__ hidden, const float* __restrict__ pma,
                               const int* __restrict__ total, float* __restrict__ summary) {
    long idx = (long)blockIdx.x * blockDim.x + threadIdx.x;
    if (idx >= (long)B * T * D) return;
    int b = idx / ((long)T * D);
    int r = idx % (T * D);
    int t = r / D, d = r % D;
    float v;
    if (t < CLS) {
        v = hidden[((long)b * S + t) * D + d];
    } else if (t < CLS + P) {
        v = pma[((long)b * P + (t - CLS)) * D + d];
    } else {
        int j = t - CLS - P;
        int shift = j + total[b] - RC;          // index into hidden[:, CLS:]
        v = (shift >= 0) ? hidden[((long)b * S + shift + CLS) * D + d] : 0.f;
    }
    summary[idx] = v;
}

// summary_mask as floats -> ws (for gating) and to the tail of d_out
__global__ void mask_kernel(const unsigned char* __restrict__ vm, const int* __restrict__ total,
                            float* __restrict__ maskf, float* __restrict__ outMask) {
    int i = blockIdx.x * blockDim.x + threadIdx.x;
    if (i >= B * T) return;
    int b = i / T, t = i % T;
    float m;
    if (t < CLS) m = vm[(long)b * S + t] ? 1.f : 0.f;
    else if (t < CLS + P) m = 1.f;
    else m = ((t - CLS - P) >= RC - total[b]) ? 1.f : 0.f;
    maskf[i] = m;
    outMask[i] = m;
}

// dual LN (shared mean/var, two gain/bias sets) over summary rows
__global__ void ln_dual_kernel(const float* __restrict__ x,
                               const float* __restrict__ g1, const float* __restrict__ b1,
                               const float* __restrict__ g2, const float* __restrict__ b2,
                               float* __restrict__ o1, float* __restrict__ o2) {
    int row = blockIdx.x;
    const float* xr = x + (long)row * D;
    float s = 0.f;
    for (int d = threadIdx.x; d < D; d += blockDim.x) s += xr[d];
    float mu = blockReduceSum(s) * (1.f / D);
    float s2 = 0.f;
    for (int d = threadIdx.x; d < D; d += blockDim.x) { float t = xr[d] - mu; s2 += t * t; }
    float var = blockReduceSum(s2) * (1.f / D);
    float rs = rsqrtf(var + 1e-5f);
    for (int d = threadIdx.x; d < D; d += blockDim.x) {
        float nx = (xr[d] - mu) * rs;
        o1[(long)row * D + d] = nx * g1[d] + b1[d];
        o2[(long)row * D + d] = nx * g2[d] + b2[d];
    }
}

// out = sigmoid(gate + b_g) * silu(val + b_v) * mask
__global__ void gate_epilogue_kernel(const float* __restrict__ valpre, const float* __restrict__ gatepre,
                                     const float* __restrict__ b_v, const float* __restrict__ b_g,
                                     const float* __restrict__ maskf, float* __restrict__ out) {
    long idx = (long)blockIdx.x * blockDim.x + threadIdx.x;
    if (idx >= (long)B * T * D) return;
    int row = idx / D, dc = idx % D;
    float v = valpre[idx] + b_v[dc];
    float g = gatepre[idx] + b_g[dc];
    float sv = v / (1.f + expf(-v));          // silu
    float sg = 1.f / (1.f + expf(-g));        // sigmoid
    out[idx] = sg * sv * maskf[row];
}

// ---------------- launcher ----------------
extern "C" void kernel_launch(void* const* d_in, const int* in_sizes, int n_in,
                              void* d_out, int out_size, void* d_ws, size_t ws_size,
                              hipStream_t stream) {
    const float* hidden    = (const float*)d_in[0];
    const unsigned char* vm = (const unsigned char*)d_in[1];   // jnp bool -> 1 byte
    const float* queries   = (const float*)d_in[2];
    const float* in_proj_w = (const float*)d_in[3];
    const float* in_proj_b = (const float*)d_in[4];
    const float* out_w     = (const float*)d_in[5];
    const float* out_b     = (const float*)d_in[6];
    const float* ln_pma_g  = (const float*)d_in[7];
    const float* ln_pma_b  = (const float*)d_in[8];
    const float* ln_v_g    = (const float*)d_in[9];
    const float* ln_v_b    = (const float*)d_in[10];
    const float* W_v       = (const float*)d_in[11];
    const float* b_v       = (const float*)d_in[12];
    const float* ln_g_g    = (const float*)d_in[13];
    const float* ln_g_b    = (const float*)d_in[14];
    const float* W_g       = (const float*)d_in[15];
    const float* b_g       = (const float*)d_in[16];
    float* out = (float*)d_out;

    // workspace layout (floats); ~51 MB total
    float* ws = (float*)d_ws;
    long o = 0;
    float* qbuf    = ws + o; o += (long)P * D;
    float* qk_eff  = ws + o; o += (long)PH * D;
    float* scores  = ws + o; o += (long)B * PH * S;     // attn in place
    float* ubuf    = ws + o; o += (long)B * PH * D;
    float* ctx     = ws + o; o += (long)B * P * D;
    float* pooled  = ws + o; o += (long)B * P * D;
    float* pma     = ws + o; o += (long)B * P * D;
    float* summary = ws + o; o += (long)B * T * D;
    float* lnv     = ws + o; o += (long)B * T * D;
    float* lng     = ws + o; o += (long)B * T * D;
    float* valpre  = ws + o; o += (long)B * T * D;
    float* gatepre = ws + o; o += (long)B * T * D;
    float* maskf   = ws + o; o += (long)B * T;
    int*   total   = (int*)(ws + o); o += B;

    // 0) q = queries @ Wq.T + bq
    qproj_kernel<<<(P * D + 255) / 256, 256, 0, stream>>>(queries, in_proj_w, in_proj_b, qbuf);
    // 1) qk_eff = per-head fold of q into Wk (kills the 68.7 GFLOP K projection)
    qkeff_kernel<<<(PH * D + 255) / 256, 256, 0, stream>>>(qbuf, in_proj_w, qk_eff);

    // 2) scores[b] (128 x S) = qk_eff (128 x D) x hidden[b]^T ; B[k=d][n=s] = hidden[b,s,d]
    wmma_gemm_f32<D, 1, 1, D, S><<<dim3(S / 256, PH / 16, B), 128, 0, stream>>>(
        qk_eff, hidden, scores, S, D,
        /*bsA*/ 0, /*bsB*/ (long)S * D, /*bsC*/ (long)PH * S);

    // 3) masked scaled softmax in place (valid_mask is a contiguous prefix)
    totals_kernel<<<B, 256, 0, stream>>>(vm, total);
    softmax_kernel<<<B * PH, 256, 0, stream>>>(scores, vm);

    // 4) u[b] (128 x D) = attn[b] (128 x S) x hidden[b] (S x D)
    wmma_gemm_f32<S, 1, D, 1, D><<<dim3(D / 256, PH / 16, B), 128, 0, stream>>>(
        scores, hidden, ubuf, D, S,
        /*bsA*/ (long)PH * S, /*bsB*/ (long)S * D, /*bsC*/ (long)PH * D);

    // 5) ctx (block-diag Wv), pooled (out_w), pma layer-norm
    ctx_kernel<<<(B * P * D + 255) / 256, 256, 0, stream>>>(ubuf, in_proj_w, in_proj_b, ctx);
    pooled_kernel<<<(B * P * D + 255) / 256, 256, 0, stream>>>(ctx, out_w, out_b, pooled);
    pma_ln_kernel<<<B * P, 256, 0, stream>>>(queries, pooled, ln_pma_g, ln_pma_b, pma);

    // 6) summary assembly + mask
    summary_kernel<<<(int)(((long)B * T * D + 255) / 256), 256, 0, stream>>>(hidden, pma, total, summary);
    mask_kernel<<<(B * T + 255) / 256, 256, 0, stream>>>(vm, total, maskf, out + (long)B * T * D);

    // 7) dual LN, then val/gate GEMMs (M = 1168 = 73*16 tiles exactly), then gating
    ln_dual_kernel<<<B * T, 256, 0, stream>>>(summary, ln_v_g, ln_v_b, ln_g_g, ln_g_b, lnv, lng);
    wmma_gemm_f32<D, 1, 1, D, D><<<dim3(D / 256, (B * T) / 16, 1), 128, 0, stream>>>(
        lnv, W_v, valpre, D, D, 0, 0, 0);
    wmma_gemm_f32<D, 1, 1, D, D><<<dim3(D / 256, (B * T) / 16, 1), 128, 0, stream>>>(
        lng, W_g, gatepre, D, D, 0, 0, 0);
    gate_epilogue_kernel<<<(int)(((long)B * T * D + 255) / 256), 256, 0, stream>>>(
        valpre, gatepre, b_v, b_g, maskf, out);

    (void)in_sizes; (void)n_in; (void)out_size; (void)ws_size;
}